// GPUAugmentations_6193342841070
// MI455X (gfx1250) — compile-verified
//
#include <hip/hip_runtime.h>
#include <hip/hip_bf16.h>

typedef __attribute__((ext_vector_type(4))) float fv4;
typedef __attribute__((ext_vector_type(4))) int   iv4;

namespace {
constexpr int Bc  = 256;
constexpr int Hc  = 224;
constexpr int Wc  = 224;
constexpr int HWc = Hc * Wc;          // 50176
constexpr int WVc = Wc / 4;           // 56 vec4 per row
constexpr int VPIc = HWc / 4;         // 12544 vec4 per channel-plane
constexpr int TPB  = 256;
constexpr int BPIc = VPIc / TPB;      // 49 blocks per image (exact)

struct __align__(16) ImgParams {
    float m[9];        // composite color matrix (includes brightness and 1/255)
    int   flip;
    int   top, bot, left, right;   // erase box, degenerate if !apply
    int   pad0, pad1;
};
static_assert(sizeof(ImgParams) == 64, "ImgParams must be 64B");

struct __align__(16) NormConst {
    float inv_std[3];
    float nms[3];      // -mean/std
    float pad[2];
};
} // namespace

__device__ __forceinline__ void mat3_mul(const float* A, const float* B, float* C) {
    #pragma unroll
    for (int i = 0; i < 3; ++i)
        #pragma unroll
        for (int j = 0; j < 3; ++j)
            C[i*3+j] = A[i*3+0]*B[0*3+j] + A[i*3+1]*B[1*3+j] + A[i*3+2]*B[2*3+j];
}

__global__ void aug_setup_kernel(
    const float* __restrict__ flip_u, const float* __restrict__ bj_u,
    const float* __restrict__ cj_u,   const float* __restrict__ sj_u,
    const float* __restrict__ hue_u,  const float* __restrict__ erase_u,
    const float* __restrict__ area_u, const float* __restrict__ aspect_u,
    const float* __restrict__ top_u,  const float* __restrict__ left_u,
    const float* __restrict__ mean,   const float* __restrict__ stdv,
    ImgParams* __restrict__ P, NormConst* __restrict__ N)
{
    const int b = blockIdx.x * blockDim.x + threadIdx.x;
    if (b == 0) {
        #pragma unroll
        for (int c = 0; c < 3; ++c) {
            float is = 1.0f / stdv[c];
            N->inv_std[c] = is;
            N->nms[c]     = -mean[c] * is;
        }
        N->pad[0] = 0.f; N->pad[1] = 0.f;
    }
    if (b >= Bc) return;

    const float b_fac = 1.0f + (bj_u[b] * 2.0f - 1.0f) * 0.4f;
    const float c_fac = 1.0f + (cj_u[b] * 2.0f - 1.0f) * 0.4f;
    const float s_fac = 1.0f + (sj_u[b] * 2.0f - 1.0f) * 0.4f;
    const float h_fac = (hue_u[b] * 2.0f - 1.0f) * 0.1f * 0.5f;
    const float ang   = h_fac * 3.14159f;
    const float ch = cosf(ang);
    const float sh = sinf(ang);

    // contrast: c*I + (1-c)/3 * J ; saturation analogous
    float Mc[9], Ms[9], Mh[9], T[9], M[9];
    const float ac = (1.0f - c_fac) / 3.0f;
    const float as = (1.0f - s_fac) / 3.0f;
    #pragma unroll
    for (int i = 0; i < 3; ++i)
        #pragma unroll
        for (int j = 0; j < 3; ++j) {
            Mc[i*3+j] = ((i == j) ? c_fac : 0.0f) + ac;
            Ms[i*3+j] = ((i == j) ? s_fac : 0.0f) + as;
        }
    // hue rotation: out_r = gray + (r-gray)ch + (g-b)*0.5*sh, cyclic
    const float a  = (1.0f - ch) / 3.0f;
    const float s2 = 0.5f * sh;
    Mh[0] = ch + a; Mh[1] = a + s2; Mh[2] = a - s2;
    Mh[3] = a - s2; Mh[4] = ch + a; Mh[5] = a + s2;
    Mh[6] = a + s2; Mh[7] = a - s2; Mh[8] = ch + a;

    mat3_mul(Ms, Mc, T);
    mat3_mul(Mh, T, M);
    const float scale = b_fac * (1.0f / 255.0f);   // fold u8 scale + brightness

    ImgParams p;
    #pragma unroll
    for (int i = 0; i < 9; ++i) p.m[i] = M[i] * scale;
    p.flip = (flip_u[b] > 0.5f) ? 1 : 0;

    const float target_area = (0.02f + area_u[b] * 0.31f) * (float)(Hc * Wc);
    const float aspect      = 0.3f + aspect_u[b] * 3.0f;
    const int h_e = (int)rintf(sqrtf(target_area * aspect));   // RTE == jnp.round
    const int w_e = (int)rintf(sqrtf(target_area / aspect));
    const bool apply = (erase_u[b] < 0.25f) && (h_e < Hc) && (w_e < Wc);
    const int top  = (int)floorf(top_u[b]  * fmaxf((float)(Hc - h_e + 1), 1.0f));
    const int left = (int)floorf(left_u[b] * fmaxf((float)(Wc - w_e + 1), 1.0f));
    if (apply) {
        p.top = top;  p.bot = top + h_e;
        p.left = left; p.right = left + w_e;
    } else {
        p.top = 0x40000000; p.bot = -0x40000000;   // empty box: row test always fails
        p.left = 0; p.right = 0;
    }
    p.pad0 = 0; p.pad1 = 0;
    P[b] = p;
}

__global__ __launch_bounds__(TPB) void aug_main_kernel(
    const int*   __restrict__ x,
    const float* __restrict__ noise,
    const ImgParams* __restrict__ P,
    const NormConst* __restrict__ N,
    float* __restrict__ out)
{
    const int b    = blockIdx.x / BPIc;               // uniform per block
    const int pix  = (blockIdx.x % BPIc) * TPB + threadIdx.x;
    const int h    = pix / WVc;
    const int col0 = (pix - h * WVc) * 4;

    const ImgParams p = P[b];                         // all lanes same address -> broadcast
    const float is0 = N->inv_std[0], is1 = N->inv_std[1], is2 = N->inv_std[2];
    const float nm0 = N->nms[0],     nm1 = N->nms[1],     nm2 = N->nms[2];

    const int plane = b * (3 * HWc) + h * Wc;         // max ~38.5M, fits int32

    // flip: output col c reads source col W-1-c; vec4 stays 16B aligned
    const int colsrc = p.flip ? (Wc - 4 - col0) : col0;
    const int* px = x + plane + colsrc;
    iv4 i0 = *(const iv4*)(px);
    iv4 i1 = *(const iv4*)(px + HWc);
    iv4 i2 = *(const iv4*)(px + 2 * HWc);
    if (p.flip) {                                      // wave-uniform branch
        i0 = i0.wzyx; i1 = i1.wzyx; i2 = i2.wzyx;
    }

    // erase-box overlap (row test folded in; degenerate box if !apply)
    const bool over = (h >= p.top) & (h < p.bot) & (col0 < p.right) & (col0 + 4 > p.left);
    const float* np = noise + plane + col0;
    if (over) {
        // gfx1250 global_prefetch_b8: warm noise lines while color math runs
        __builtin_prefetch(np,            0, 3);
        __builtin_prefetch(np + HWc,      0, 3);
        __builtin_prefetch(np + 2 * HWc,  0, 3);
    }

    const float m0 = p.m[0], m1 = p.m[1], m2 = p.m[2];
    const float m3 = p.m[3], m4 = p.m[4], m5 = p.m[5];
    const float m6 = p.m[6], m7 = p.m[7], m8 = p.m[8];

    fv4 o0, o1, o2;
    #pragma unroll
    for (int j = 0; j < 4; ++j) {
        const float rr = (float)i0[j];
        const float gg = (float)i1[j];
        const float bb = (float)i2[j];
        float t0 = fmaf(m2, bb, fmaf(m1, gg, m0 * rr));
        float t1 = fmaf(m5, bb, fmaf(m4, gg, m3 * rr));
        float t2 = fmaf(m8, bb, fmaf(m7, gg, m6 * rr));
        t0 = fminf(fmaxf(t0, 0.0f), 1.0f);             // -> v_med3
        t1 = fminf(fmaxf(t1, 0.0f), 1.0f);
        t2 = fminf(fmaxf(t2, 0.0f), 1.0f);
        o0[j] = fmaf(t0, is0, nm0);
        o1[j] = fmaf(t1, is1, nm1);
        o2[j] = fmaf(t2, is2, nm2);
    }

    if (over) {                                        // rare: ~4-5% of pixels
        const fv4 n0 = *(const fv4*)(np);
        const fv4 n1 = *(const fv4*)(np + HWc);
        const fv4 n2 = *(const fv4*)(np + 2 * HWc);
        #pragma unroll
        for (int j = 0; j < 4; ++j) {
            const int c = col0 + j;
            if ((c >= p.left) & (c < p.right)) {
                o0[j] = n0[j]; o1[j] = n1[j]; o2[j] = n2[j];
            }
        }
    }

    float* op = out + plane + col0;                    // stream-once: NT stores
    __builtin_nontemporal_store(o0, (fv4*)(op));
    __builtin_nontemporal_store(o1, (fv4*)(op + HWc));
    __builtin_nontemporal_store(o2, (fv4*)(op + 2 * HWc));
}

extern "C" void kernel_launch(void* const* d_in, const int* in_sizes, int n_in,
                              void* d_out, int out_size, void* d_ws, size_t ws_size,
                              hipStream_t stream) {
    const int*   x        = (const int*)  d_in[0];
    const float* flip_u   = (const float*)d_in[1];
    const float* bj_u     = (const float*)d_in[2];
    const float* cj_u     = (const float*)d_in[3];
    const float* sj_u     = (const float*)d_in[4];
    const float* hue_u    = (const float*)d_in[5];
    const float* erase_u  = (const float*)d_in[6];
    const float* area_u   = (const float*)d_in[7];
    const float* aspect_u = (const float*)d_in[8];
    const float* top_u    = (const float*)d_in[9];
    const float* left_u   = (const float*)d_in[10];
    const float* noise    = (const float*)d_in[11];
    const float* mean     = (const float*)d_in[12];
    const float* stdv     = (const float*)d_in[13];
    float* out = (float*)d_out;

    ImgParams* P = (ImgParams*)d_ws;                       // 256 * 64B
    NormConst* N = (NormConst*)((char*)d_ws + Bc * sizeof(ImgParams));

    aug_setup_kernel<<<1, TPB, 0, stream>>>(
        flip_u, bj_u, cj_u, sj_u, hue_u, erase_u, area_u, aspect_u,
        top_u, left_u, mean, stdv, P, N);

    aug_main_kernel<<<Bc * BPIc, TPB, 0, stream>>>(x, noise, P, N, out);
}